// Network_23441931502118
// MI455X (gfx1250) — compile-verified
//
#include <hip/hip_runtime.h>
#include <stdint.h>
#include <stddef.h>

// ---------------- sizes / params (match reference) ----------------
namespace {
constexpr int kT    = 256;
constexpr int kNIn  = 2048;
constexpr int kNE   = 4096;
constexpr int kNI   = 1024;
constexpr int kPreE = kNIn + kNE + kNI;  // 7168
constexpr int kPreI = kNE + kNI;         // 5120

constexpr float kDt         = 0.001f;
constexpr float kTauSynInv  = 200.0f;
constexpr float kTauMemInv  = 100.0f;
constexpr float kVTh        = 1.0f;
constexpr float kRhoReset   = 5.0f;
constexpr float kTauPreInv  = 20.0f;
constexpr float kTauPostInv = 20.0f;
constexpr float kEtaPlus    = 1e-3f;
constexpr float kEtaMinus   = 1e-3f;

constexpr int kSegCols = 1024;           // columns per block (8 waves)
constexpr int kGyE = kPreE / kSegCols;   // 7
constexpr int kGyI = kPreI / kSegCols;   // 5
}  // namespace

typedef float v2f __attribute__((ext_vector_type(2)));
typedef float v8f __attribute__((ext_vector_type(8)));

// ---------------------------------------------------------------------------
// Fused kernel: apply previous step's STDP delta to transposed masked weights
// (clip to [0,1], re-mask from bit mask), then accumulate the mat-vec partial
// with V_WMMA_F32_16X16X4_F32.  wT layout: wT[c*N + r] (column-major of w).
// Grid: (N/16, P/segCols), block 256 = 8 waves. Each wave: 16 rows x
// (segCols/8) columns, K-chunks of 4.  The K loop uses a uniform induction
// variable so EXEC stays all-ones at every WMMA (ISA requirement).
// ---------------------------------------------------------------------------
__global__ __launch_bounds__(256) void fused_stdp_matvec(
    float* __restrict__ wT, const uint32_t* __restrict__ mbits,
    const float* __restrict__ pre_new,   // pre vector for THIS step's matvec
    const float* __restrict__ pre_old,   // pre vector of previous step (z_pre)
    const float* __restrict__ z_post,    // previous step's post spikes
    const float* __restrict__ t_pre,     // traces after previous step's update
    const float* __restrict__ t_post,
    float* __restrict__ partial,         // [gridDim.y][N] partial currents
    int N, int segCols) {
  __shared__ float red[8][16];

  const int lane = threadIdx.x & 31;
  const int wid  = threadIdx.x >> 5;
  const int n    = lane & 15;   // row-in-tile (B matrix N index)
  const int hi   = lane >> 4;   // half-wave: K pair select
  const int r0   = blockIdx.x << 4;
  const int colsPerWave = segCols >> 3;          // uniform
  const int cBase = blockIdx.y * segCols + wid * colsPerWave;
  const int nWords  = N >> 5;
  const int wordRow = r0 >> 5;
  const int bit     = (r0 & 31) + n;  // r0 multiple of 16 -> bit in [0,31]

  // per-row STDP coefficients: delta(c,r) = ap*t_pre[c] - am*pre_old[c]
  const float ap = kEtaPlus  * z_post[r0 + n];
  const float am = kEtaMinus * t_post[r0 + n];

  v8f acc = {};

#pragma unroll 2
  for (int k = 0; k < colsPerWave; k += 4) {     // uniform trip count
    const int cl = cBase + k + (hi << 1);        // this lane's first column
    const size_t base = (size_t)cl * N + (size_t)(r0 + n);

    float w0 = wT[base];
    float w1 = wT[base + N];
    const uint32_t mw0 = mbits[(size_t)cl * nWords + wordRow];
    const uint32_t mw1 = mbits[(size_t)(cl + 1) * nWords + wordRow];

    const float2 tp = *(const float2*)(t_pre + cl);    // cl is even, 8B aligned
    const float2 po = *(const float2*)(pre_old + cl);
    const float2 pn = *(const float2*)(pre_new + cl);

    const float u0 = fminf(fmaxf(w0 + ap * tp.x - am * po.x, 0.0f), 1.0f);
    const float u1 = fminf(fmaxf(w1 + ap * tp.y - am * po.y, 0.0f), 1.0f);
    w0 = ((mw0 >> bit) & 1u) ? u0 : 0.0f;
    w1 = ((mw1 >> bit) & 1u) ? u1 : 0.0f;
    wT[base]     = w0;
    wT[base + N] = w1;

    // stream ahead: next 16 columns of this row block
    __builtin_prefetch((const void*)(wT + base + (size_t)16 * (size_t)N), 0, 1);

    // A: pre vector broadcast over all 16 M rows; B: updated weight tile.
    v2f a;  a.x = pn.x;  a.y = pn.y;
    v2f b;  b.x = w0;    b.y = w1;
    acc = __builtin_amdgcn_wmma_f32_16x16x4_f32(
        false, a, false, b, (short)0, acc, false, false);
  }

  // Every C row is identical (A rows identical); lane n (hi==0), VGPR0 = C[0][n].
  if (hi == 0) red[wid][n] = acc[0];
  __syncthreads();
  if (threadIdx.x < 16) {
    float s = 0.0f;
#pragma unroll
    for (int w = 0; w < 8; ++w) s += red[w][threadIdx.x];
    partial[(size_t)blockIdx.y * N + r0 + threadIdx.x] = s;
  }
}

// ---------------------------------------------------------------------------
// Excitatory LIF + trace update. Also builds pre_i(t) = [ze(t), -zi(t-1)].
// Grid covers kPreE threads.
// ---------------------------------------------------------------------------
__global__ void lif_exc_kernel(
    const float* __restrict__ partial, int gy,
    float* __restrict__ ze, float* __restrict__ ve,
    float* __restrict__ ie, float* __restrict__ re,
    float* __restrict__ tq_e, float* __restrict__ tp_e,
    const float* __restrict__ preE_cur, const float* __restrict__ zi_old,
    float* __restrict__ preI_cur, float* __restrict__ out_row) {
  const int j = blockIdx.x * blockDim.x + threadIdx.x;
  if (j < kNE) {
    float cur = 0.0f;
    for (int s = 0; s < gy; ++s) cur += partial[(size_t)s * kNE + j];
    const float v = ve[j], i = ie[j], rho = re[j];
    const float refrac = (rho > 0.0f) ? 1.0f : 0.0f;
    const float v_dec = v + kDt * kTauMemInv * ((0.0f - v) + i);
    const float i_dec = i - kDt * kTauSynInv * i;
    float z = (v_dec - kVTh > 0.0f) ? 1.0f : 0.0f;
    float v_new = (1.0f - z) * v_dec;          // V_RESET = 0
    const float i_new = i_dec + cur;
    v_new = (1.0f - refrac) * v_new + refrac * v;
    z = (1.0f - refrac) * z;
    const float rho_new = (1.0f - z) * fmaxf(rho - refrac, 0.0f) + z * kRhoReset;
    ze[j] = z; ve[j] = v_new; ie[j] = i_new; re[j] = rho_new;
    out_row[j] = z;
    const float tq = tq_e[j];
    tq_e[j] = tq + kDt * kTauPostInv * (-tq + z);   // A_POST = 1
    preI_cur[j] = z;                                 // pre_i[0:NE] = ze(t)
  }
  if (j >= kNE && j < kPreI) preI_cur[j] = -zi_old[j - kNE];
  if (j < kPreE) {
    const float tp = tp_e[j];
    tp_e[j] = tp + kDt * kTauPreInv * (-tp + preE_cur[j]);  // A_PRE = 1
  }
}

// ---------------------------------------------------------------------------
// Inhibitory LIF + trace update. Grid covers kPreI threads.
// ---------------------------------------------------------------------------
__global__ void lif_inh_kernel(
    const float* __restrict__ partial, int gy,
    float* __restrict__ zi, float* __restrict__ vi,
    float* __restrict__ ii, float* __restrict__ ri,
    float* __restrict__ tq_i, float* __restrict__ tp_i,
    const float* __restrict__ preI_cur) {
  const int j = blockIdx.x * blockDim.x + threadIdx.x;
  if (j < kNI) {
    float cur = 0.0f;
    for (int s = 0; s < gy; ++s) cur += partial[(size_t)s * kNI + j];
    const float v = vi[j], i = ii[j], rho = ri[j];
    const float refrac = (rho > 0.0f) ? 1.0f : 0.0f;
    const float v_dec = v + kDt * kTauMemInv * ((0.0f - v) + i);
    const float i_dec = i - kDt * kTauSynInv * i;
    float z = (v_dec - kVTh > 0.0f) ? 1.0f : 0.0f;
    float v_new = (1.0f - z) * v_dec;
    const float i_new = i_dec + cur;
    v_new = (1.0f - refrac) * v_new + refrac * v;
    z = (1.0f - refrac) * z;
    const float rho_new = (1.0f - z) * fmaxf(rho - refrac, 0.0f) + z * kRhoReset;
    zi[j] = z; vi[j] = v_new; ii[j] = i_new; ri[j] = rho_new;
    const float tq = tq_i[j];
    tq_i[j] = tq + kDt * kTauPostInv * (-tq + z);
  }
  if (j < kPreI) {
    const float tp = tp_i[j];
    tp_i[j] = tp + kDt * kTauPreInv * (-tp + preI_cur[j]);
  }
}

// Build pre_e(t+1) = [x_{t+1}, ze(t), -zi(t)].
__global__ void build_preE_kernel(const float* __restrict__ x_next,
                                  const float* __restrict__ ze,
                                  const float* __restrict__ zi,
                                  float* __restrict__ preE_next) {
  const int j = blockIdx.x * blockDim.x + threadIdx.x;
  if (j >= kPreE) return;
  float v;
  if (j < kNIn)            v = x_next[j];
  else if (j < kNIn + kNE) v = ze[j - kNIn];
  else                     v = -zi[j - kNIn - kNE];
  preE_next[j] = v;
}

// ---------------------------------------------------------------------------
// One-time per launch: transpose masked weights to column-major workspace and
// bit-pack the mask (32 rows per word along r).
// ---------------------------------------------------------------------------
__global__ void pack_w_kernel(const float* __restrict__ w,
                              const float* __restrict__ mask,
                              float* __restrict__ wT,
                              uint32_t* __restrict__ mbits, int N, int P) {
  const int nWords = N >> 5;
  const int idx = blockIdx.x * blockDim.x + threadIdx.x;
  if (idx >= P * nWords) return;
  const int c  = idx / nWords;
  const int rw = idx - c * nWords;
  uint32_t bits = 0;
  const size_t obase = (size_t)c * N + ((size_t)rw << 5);
  for (int jj = 0; jj < 32; ++jj) {
    const size_t r = ((size_t)rw << 5) + jj;
    const float m = mask[r * P + c];
    const float wv = w[r * P + c];
    bits |= (uint32_t)(m != 0.0f) << jj;
    wT[obase + jj] = wv * m;   // masked weight (mask is 0/1)
  }
  mbits[idx] = bits;
}

// Zero the small state region; seed pre_e(0) = [x_0, 0, 0].
__global__ void init_state_kernel(float* __restrict__ S, int total,
                                  int bufE0_off, const float* __restrict__ x0) {
  const int idx = blockIdx.x * blockDim.x + threadIdx.x;
  if (idx >= total) return;
  float v = 0.0f;
  const int d = idx - bufE0_off;
  if (d >= 0 && d < kNIn) v = x0[d];
  S[idx] = v;
}

// ---------------------------------------------------------------------------
extern "C" void kernel_launch(void* const* d_in, const int* in_sizes, int n_in,
                              void* d_out, int out_size, void* d_ws,
                              size_t ws_size, hipStream_t stream) {
  (void)in_sizes; (void)n_in; (void)out_size; (void)ws_size;
  const float* x      = (const float*)d_in[0];  // [T, NIn]
  const float* w_e    = (const float*)d_in[1];  // [NE, PreE]
  const float* w_i    = (const float*)d_in[2];  // [NI, PreI]
  const float* mask_e = (const float*)d_in[3];
  const float* mask_i = (const float*)d_in[4];
  float* out = (float*)d_out;                   // [T, NE]

  // ---- workspace carve ----
  char* base = (char*)d_ws;
  size_t off = 0;
  auto take = [&](size_t bytes) -> void* {
    void* p = base + off;
    off += (bytes + 255) & ~(size_t)255;
    return p;
  };
  float*    wTe = (float*)take(sizeof(float) * (size_t)kPreE * kNE);
  float*    wTi = (float*)take(sizeof(float) * (size_t)kPreI * kNI);
  uint32_t* mTe = (uint32_t*)take(sizeof(uint32_t) * (size_t)kPreE * (kNE / 32));
  uint32_t* mTi = (uint32_t*)take(sizeof(uint32_t) * (size_t)kPreI * (kNI / 32));

  float* S = (float*)(base + off);
  size_t so = 0;
  auto ftake = [&](size_t count) -> float* { float* p = S + so; so += count; return p; };
  float* partE = ftake((size_t)kGyE * kNE);
  float* partI = ftake((size_t)kGyI * kNI);
  float* ze = ftake(kNE); float* ve = ftake(kNE);
  float* ie = ftake(kNE); float* re = ftake(kNE);
  float* zi = ftake(kNI); float* vi = ftake(kNI);
  float* ii = ftake(kNI); float* ri = ftake(kNI);
  float* tpe = ftake(kPreE); float* tqe = ftake(kNE);
  float* tpi = ftake(kPreI); float* tqi = ftake(kNI);
  const int bufE0_off = (int)so;
  float* bufE[2]; bufE[0] = ftake(kPreE); bufE[1] = ftake(kPreE);
  float* bufI[2]; bufI[0] = ftake(kPreI); bufI[1] = ftake(kPreI);
  const int totalS = (int)so;

  // ---- one-time init ----
  pack_w_kernel<<<(kPreE * (kNE / 32) + 255) / 256, 256, 0, stream>>>(
      w_e, mask_e, wTe, mTe, kNE, kPreE);
  pack_w_kernel<<<(kPreI * (kNI / 32) + 255) / 256, 256, 0, stream>>>(
      w_i, mask_i, wTi, mTi, kNI, kPreI);
  init_state_kernel<<<(totalS + 255) / 256, 256, 0, stream>>>(
      S, totalS, bufE0_off, x);

  // ---- time loop ----
  for (int t = 0; t < kT; ++t) {
    const float* preE_cur = bufE[t & 1];
    const float* preE_old = bufE[(t + 1) & 1];
    float*       preI_cur = bufI[t & 1];
    const float* preI_old = bufI[(t + 1) & 1];

    fused_stdp_matvec<<<dim3(kNE / 16, kGyE), 256, 0, stream>>>(
        wTe, mTe, preE_cur, preE_old, ze, tpe, tqe, partE, kNE, kSegCols);
    lif_exc_kernel<<<(kPreE + 255) / 256, 256, 0, stream>>>(
        partE, kGyE, ze, ve, ie, re, tqe, tpe, preE_cur, zi, preI_cur,
        out + (size_t)t * kNE);
    fused_stdp_matvec<<<dim3(kNI / 16, kGyI), 256, 0, stream>>>(
        wTi, mTi, preI_cur, preI_old, zi, tpi, tqi, partI, kNI, kSegCols);
    lif_inh_kernel<<<(kPreI + 255) / 256, 256, 0, stream>>>(
        partI, kGyI, zi, vi, ii, ri, tqi, tpi, preI_cur);
    if (t + 1 < kT) {
      build_preE_kernel<<<(kPreE + 255) / 256, 256, 0, stream>>>(
          x + (size_t)(t + 1) * kNIn, ze, zi, bufE[(t + 1) & 1]);
    }
  }
}